// LayerwiseBitPredictor_68496138437351
// MI455X (gfx1250) — compile-verified
//
#include <hip/hip_runtime.h>
#include <hip/hip_bf16.h>

typedef __attribute__((ext_vector_type(2))) float v2f;
typedef __attribute__((ext_vector_type(4))) float v4f;
typedef __attribute__((ext_vector_type(8))) float v8f;

#define BB 64
#define SS 2048
#define HH 1024
#define PP 8
#define NCHUNK 16
#define SCHUNK (SS / NCHUNK) /* 128 */

// ---------------------------------------------------------------------------
// Kernel 1: streaming partial mean-pool over S. 512 MB of NT float4 loads.
// grid = BB*NCHUNK blocks x 256 threads; each thread owns 4 h-columns.
// ---------------------------------------------------------------------------
__global__ __launch_bounds__(256) void pool_partial_kernel(
    const float* __restrict__ hs, float* __restrict__ partial) {
  const int blk = blockIdx.x;            // b * NCHUNK + c
  const int b   = blk / NCHUNK;
  const int c   = blk % NCHUNK;
  const int t   = threadIdx.x;           // h = 4*t .. 4*t+3

  const v4f* src =
      (const v4f*)(hs + ((size_t)b * SS + (size_t)c * SCHUNK) * HH) + t;
  v4f acc = {0.f, 0.f, 0.f, 0.f};
#pragma unroll 4
  for (int s = 0; s < SCHUNK; ++s) {
    v4f v = __builtin_nontemporal_load(src);
    acc += v;
    src += HH / 4;                       // next sequence position (4 KB stride)
  }
  *((v4f*)(partial + (size_t)blk * HH) + t) = acc;
}

// ---------------------------------------------------------------------------
// Kernel 2: fold NCHUNK partials, scale by 1/S -> pooled (B,H)
// ---------------------------------------------------------------------------
__global__ __launch_bounds__(256) void pool_reduce_kernel(
    const float* __restrict__ partial, float* __restrict__ pooled) {
  const int b = blockIdx.x;
  const int t = threadIdx.x;
  v4f acc = {0.f, 0.f, 0.f, 0.f};
#pragma unroll
  for (int c = 0; c < NCHUNK; ++c) {
    acc += *((const v4f*)(partial + (size_t)(b * NCHUNK + c) * HH) + t);
  }
  const float inv = 1.0f / (float)SS;
  acc *= inv;
  *((v4f*)(pooled + (size_t)b * HH) + t) = acc;
}

// ---------------------------------------------------------------------------
// Kernel 3: h = silu(pooled @ W1.T + b1) via V_WMMA_F32_16X16X4_F32.
// One wave per 16x16 output tile; K loop in steps of 4.
//   A (16x4 f32):  lanes 0-15 m=0..15 hold K={k,k+1}; lanes 16-31 hold K={k+2,k+3}
//   B (4x16 f32):  lane n=0..15 holds rows K={k,k+1} / lanes 16-31 K={k+2,k+3}
//                  B[kk][n] = W1[n][kk]   (W1 row-major (out,in))
//   C/D (16x16):   VGPR v -> M=v (lanes 0-15), M=v+8 (lanes 16-31), N=lane&15
// ---------------------------------------------------------------------------
__global__ __launch_bounds__(32) void gemm1_silu_wmma_kernel(
    const float* __restrict__ pooled, const float* __restrict__ W1,
    const float* __restrict__ b1, float* __restrict__ hout) {
  const int tile = blockIdx.x;           // 0..255
  const int tm   = (tile & 3) * 16;      // batch-tile (64 rows -> 4 tiles)
  const int tn   = (tile >> 2) * 16;     // output-feature tile (1024 -> 64)
  const int lane = threadIdx.x;
  const int half = lane >> 4;            // 0: K pair {0,1}, 1: K pair {2,3}
  const int r    = lane & 15;

  const float* arow = pooled + (size_t)(tm + r) * HH + 2 * half;
  const float* brow = W1 + (size_t)(tn + r) * HH + 2 * half;   // W1[n][k]

  v8f acc = {};
#pragma unroll 2
  for (int k = 0; k < HH; k += 4) {
    v2f a = *(const v2f*)(arow + k);
    v2f bmat = *(const v2f*)(brow + k);
    acc = __builtin_amdgcn_wmma_f32_16x16x4_f32(
        /*neg_a=*/false, a, /*neg_b=*/false, bmat,
        /*c_mod=*/(short)0, acc, /*reuse_a=*/false, /*reuse_b=*/false);
  }

  const float bias = b1[tn + r];         // bias depends on column n only
#pragma unroll
  for (int v = 0; v < 8; ++v) {
    const int m = tm + v + 8 * half;
    const float x = acc[v] + bias;
    const float sig = 1.0f / (1.0f + expf(-x));
    hout[(size_t)m * HH + tn + r] = x * sig;
  }
}

// ---------------------------------------------------------------------------
// Kernel 4: logits = h @ W2.T + b2; softmax(logits/temp); bit allocation.
// One wave per batch row; butterfly shuffle reduction for the 8 dot products.
// ---------------------------------------------------------------------------
__global__ __launch_bounds__(32) void head_kernel(
    const float* __restrict__ hbuf, const float* __restrict__ W2,
    const float* __restrict__ b2, const float* __restrict__ temperature,
    const int* __restrict__ total_budget, float* __restrict__ out) {
  const int b = blockIdx.x;
  const int lane = threadIdx.x;

  float acc[PP];
#pragma unroll
  for (int p = 0; p < PP; ++p) acc[p] = 0.f;

  for (int i = lane; i < HH; i += 32) {
    const float hv = hbuf[(size_t)b * HH + i];
#pragma unroll
    for (int p = 0; p < PP; ++p) acc[p] += hv * W2[(size_t)p * HH + i];
  }
#pragma unroll
  for (int p = 0; p < PP; ++p) {
#pragma unroll
    for (int off = 16; off > 0; off >>= 1) acc[p] += __shfl_xor(acc[p], off, 32);
  }

  if (lane == 0) {
    const float budget = (float)total_budget[0];
    const float temp = fminf(fmaxf(temperature[0], 0.1f), 5.0f);

    float y[PP], probs[PP], x[PP], bits[PP];
    float mx = -INFINITY;
#pragma unroll
    for (int p = 0; p < PP; ++p) {
      y[p] = (acc[p] + b2[p]) / temp;
      mx = fmaxf(mx, y[p]);
    }
    float se = 0.f;
#pragma unroll
    for (int p = 0; p < PP; ++p) { probs[p] = expf(y[p] - mx); se += probs[p]; }
    const float invse = 1.0f / se;
    float sx = 0.f;
#pragma unroll
    for (int p = 0; p < PP; ++p) {
      probs[p] *= invse;
      x[p] = fminf(fmaxf(probs[p] * budget, 2.0f), 8.0f);
      sx += x[p];
    }
    const float scale = budget / (sx + 1e-8f);
    float sbits = 0.f;
#pragma unroll
    for (int p = 0; p < PP; ++p) {
      x[p] *= scale;
      bits[p] = rintf(x[p]);             // round-half-even == jnp.round
      sbits += bits[p];
    }
    const float diff = truncf(budget - sbits);
    if (diff != 0.0f) {
      int idx = 0;
      float best = bits[0];
      if (diff > 0.0f) {
        for (int p = 1; p < PP; ++p)
          if (bits[p] > best) { best = bits[p]; idx = p; }   // first argmax
      } else {
        for (int p = 1; p < PP; ++p)
          if (bits[p] < best) { best = bits[p]; idx = p; }   // first argmin
      }
      bits[idx] = fminf(fmaxf(bits[idx] + diff, 2.0f), 8.0f);
    }
#pragma unroll
    for (int p = 0; p < PP; ++p) {
      out[b * PP + p] = bits[p];                   // bits first
      out[BB * PP + b * PP + p] = probs[p];        // then attn_probs
    }
  }
}

// ---------------------------------------------------------------------------
extern "C" void kernel_launch(void* const* d_in, const int* in_sizes, int n_in,
                              void* d_out, int out_size, void* d_ws,
                              size_t ws_size, hipStream_t stream) {
  const float* hs  = (const float*)d_in[0];   // (B,S,H)
  const float* W1  = (const float*)d_in[1];   // (H,H) row-major (out,in)
  const float* b1  = (const float*)d_in[2];   // (H)
  const float* W2  = (const float*)d_in[3];   // (P,H)
  const float* b2  = (const float*)d_in[4];   // (P)
  const float* tmp = (const float*)d_in[5];   // (1)
  const int* budget = (const int*)d_in[6];    // scalar int

  float* ws      = (float*)d_ws;
  float* partial = ws;                                   // B*NCHUNK*H floats
  float* pooled  = partial + (size_t)BB * NCHUNK * HH;   // B*H floats
  float* hbuf    = pooled + (size_t)BB * HH;             // B*H floats
  float* outf    = (float*)d_out;

  pool_partial_kernel<<<BB * NCHUNK, 256, 0, stream>>>(hs, partial);
  pool_reduce_kernel<<<BB, 256, 0, stream>>>(partial, pooled);
  gemm1_silu_wmma_kernel<<<(BB / 16) * (HH / 16), 32, 0, stream>>>(pooled, W1,
                                                                   b1, hbuf);
  head_kernel<<<BB, 32, 0, stream>>>(hbuf, W2, b2, tmp, budget, outf);
}